// SwitchRouter_69432441307480
// MI455X (gfx1250) — compile-verified
//
#include <hip/hip_runtime.h>
#include <hip/hip_bf16.h>

typedef __attribute__((ext_vector_type(2))) float v2f;
typedef __attribute__((ext_vector_type(8))) float v8f;

#define HID 2048
#define NEXP 8
#define NTOK 32768            // 8 * 4096
#define CAP 512
#define TOK_PER_BLOCK 128
#define NBLOCKS 256           // NTOK / TOK_PER_BLOCK
#define KCHUNK 1024           // K tile resident in LDS (2 chunks)

// ---------------------------------------------------------------------------
// Kernel 1: router GEMM (WMMA f32 16x16x4) + softmax/argmax + per-block stats
// ---------------------------------------------------------------------------
__global__ __launch_bounds__(256) void router_gemm_kernel(
    const float* __restrict__ hidden,   // [NTOK, HID]
    const float* __restrict__ W,        // [HID, NEXP]
    float* __restrict__ probs_out,      // [NTOK, NEXP]
    int*   __restrict__ expIdx,         // [NTOK]
    int*   __restrict__ rank,           // [NTOK]
    float* __restrict__ weight,         // [NTOK]
    int*   __restrict__ blockCounts,    // [NBLOCKS, NEXP]
    float* __restrict__ partialAux,     // [NBLOCKS]
    float* __restrict__ partialZ)       // [NBLOCKS]
{
    __shared__ float Wt[NEXP * KCHUNK];     // W^T chunk: Wt[n][k] 32 KB
    __shared__ int   sIdx[TOK_PER_BLOCK];
    __shared__ float sAcc[2];

    const int tid  = threadIdx.x;
    const int wave = tid >> 5;              // 0..7
    const int lane = tid & 31;
    const int dlt  = lane >> 4;             // K half-select (0/1)
    const int n    = lane & 15;             // B/C column (expert); A row (token)

    const int blockTokenBase = blockIdx.x * TOK_PER_BLOCK;
    const int waveTokenBase  = blockTokenBase + wave * 16;

    if (tid < 2) sAcc[tid] = 0.0f;

    const float* aRow = hidden + (size_t)(waveTokenBase + n) * HID + 2 * dlt;
    const float* bRow = Wt + (n & 7) * KCHUNK + 2 * dlt;   // n>=8 aliases, harmless

    // fill: linear coalesced global reads; transpose happens on LDS-store side
    const int fillN = tid & 7;              // expert column of W element
    const int fillK = tid >> 3;             // k row (step 32)
    float* fillDst = Wt + fillN * KCHUNK + fillK;

    v8f c = {};
    for (int kc = 0; kc < HID / KCHUNK; ++kc) {
        __syncthreads();  // protect Wt from lagging readers / first-time sAcc init
        {
            const float* src = W + kc * (KCHUNK * NEXP) + tid;   // linear stream
            #pragma unroll
            for (int it = 0; it < (KCHUNK * NEXP) / 256; ++it) {
                fillDst[it * 32] = src[it * 256];                // k advances by 32
            }
        }
        __syncthreads();

        const float* aP = aRow + kc * KCHUNK;
        const float* bP = bRow;
        #pragma unroll 8
        for (int ks = 0; ks < KCHUNK / 4; ++ks) {
            v2f a = *(const v2f*)(aP + ks * 4);   // h[tok][k0+2d .. +1]
            v2f b = *(const v2f*)(bP + ks * 4);   // W [k0+2d .. +1][n]
            c = __builtin_amdgcn_wmma_f32_16x16x4_f32(
                    false, a, false, b, (short)0, c, false, false);
        }
    }

    // ---- per-token softmax / argmax over the 8 expert lanes -----------------
    float auxAcc = 0.0f, zAcc = 0.0f;
    #pragma unroll
    for (int r = 0; r < 8; ++r) {
        float v   = c[r];
        int   tok = waveTokenBase + r + 8 * dlt;

        float mx = v; int mi = n;
        #pragma unroll
        for (int mask = 1; mask < 8; mask <<= 1) {
            float ov = __shfl_xor(mx, mask, 32);
            int   oi = __shfl_xor(mi, mask, 32);
            if (ov > mx || (ov == mx && oi < mi)) { mx = ov; mi = oi; }
        }
        float ex = __expf(v - mx);
        float s  = ex;
        #pragma unroll
        for (int mask = 1; mask < 8; mask <<= 1) s += __shfl_xor(s, mask, 32);

        float p = ex / s;
        if (n < 8) probs_out[(size_t)tok * NEXP + n] = p;

        float q = p * p;
        #pragma unroll
        for (int mask = 1; mask < 8; mask <<= 1) q += __shfl_xor(q, mask, 32);

        if (n == 0) {
            weight[tok] = 1.0f / s;            // max prob = exp(mx-mx)/s
            expIdx[tok] = mi;
            sIdx[tok - blockTokenBase] = mi;
            float lse = mx + __logf(s);
            auxAcc += q;                       // sum_e p^2 for this token
            zAcc   += lse * lse;
        }
    }
    if (n == 0) { atomicAdd(&sAcc[0], auxAcc); atomicAdd(&sAcc[1], zAcc); }
    __syncthreads();

    // ---- intra-block ranks + per-block expert histogram ---------------------
    if (tid < TOK_PER_BLOCK) {
        int e = sIdx[tid], r = 0;
        for (int j = 0; j < tid; ++j) r += (sIdx[j] == e);
        rank[blockTokenBase + tid] = r;
    }
    if (tid >= TOK_PER_BLOCK && tid < TOK_PER_BLOCK + NEXP) {
        int e = tid - TOK_PER_BLOCK, cnt = 0;
        for (int j = 0; j < TOK_PER_BLOCK; ++j) cnt += (sIdx[j] == e);
        blockCounts[blockIdx.x * NEXP + e] = cnt;
    }
    if (tid == 0) {
        partialAux[blockIdx.x] = sAcc[0];
        partialZ[blockIdx.x]   = sAcc[1];
    }
}

// ---------------------------------------------------------------------------
// Kernel 2: exclusive scan of block histograms per (batch,expert) + losses
// ---------------------------------------------------------------------------
__global__ __launch_bounds__(128) void finalize_kernel(
    const int* __restrict__ blockCounts, int* __restrict__ blockOffsets,
    const float* __restrict__ partialAux, const float* __restrict__ partialZ,
    float* __restrict__ out_aux, float* __restrict__ out_z)
{
    int t = threadIdx.x;
    if (t < 64) {                       // (b,e) pair; 32 blocks per batch row
        int b = t >> 3, e = t & 7, run = 0;
        for (int blk = 0; blk < 32; ++blk) {
            int g = b * 32 + blk;
            blockOffsets[g * NEXP + e] = run;
            run += blockCounts[g * NEXP + e];
        }
    } else if (t == 64) {
        float s = 0.0f;
        for (int i = 0; i < NBLOCKS; ++i) s += partialAux[i];
        *out_aux = s * ((float)NEXP / (float)NTOK);
    } else if (t == 65) {
        float s = 0.0f;
        for (int i = 0; i < NBLOCKS; ++i) s += partialZ[i];
        *out_z = s / (float)NTOK;
    }
}

// ---------------------------------------------------------------------------
// Kernel 3: emit dispatch / combine tensors
// ---------------------------------------------------------------------------
__global__ __launch_bounds__(256) void scatter_kernel(
    const int* __restrict__ expIdx, const int* __restrict__ rank,
    const float* __restrict__ weight, const int* __restrict__ blockOffsets,
    float* __restrict__ dispatch, float* __restrict__ combine)
{
    int t = blockIdx.x * 256 + threadIdx.x;       // token id
    int e = expIdx[t];
    int pos = blockOffsets[(t >> 7) * NEXP + e] + rank[t];
    float keep = (pos < CAP) ? 1.0f : 0.0f;
    float w = weight[t] * keep;
    #pragma unroll
    for (int j = 0; j < NEXP; ++j) {
        float d = (j == e) ? keep : 0.0f;
        dispatch[(size_t)t * NEXP + j] = d;
        combine [(size_t)t * NEXP + j] = (j == e) ? w : 0.0f;
    }
}

// ---------------------------------------------------------------------------
extern "C" void kernel_launch(void* const* d_in, const int* in_sizes, int n_in,
                              void* d_out, int out_size, void* d_ws, size_t ws_size,
                              hipStream_t stream) {
    const float* hidden = (const float*)d_in[0];   // [8,4096,2048] f32
    const float* W      = (const float*)d_in[1];   // [2048,8] f32

    float* out      = (float*)d_out;
    float* dispatch = out;                          // [B,S,E] = 262144
    float* combine  = out + 262144;
    float* probs    = out + 524288;
    float* aux      = out + 786432;
    float* z        = out + 786433;

    char* ws = (char*)d_ws;
    int*   expIdx       = (int*)  (ws);                 // 131072 B
    int*   rnk          = (int*)  (ws + 131072);        // 131072 B
    float* wgt          = (float*)(ws + 262144);        // 131072 B
    int*   blockCounts  = (int*)  (ws + 393216);        //   8192 B
    int*   blockOffsets = (int*)  (ws + 401408);        //   8192 B
    float* partialAux   = (float*)(ws + 409600);        //   1024 B
    float* partialZ     = (float*)(ws + 410624);        //   1024 B

    router_gemm_kernel<<<NBLOCKS, 256, 0, stream>>>(
        hidden, W, probs, expIdx, rnk, wgt, blockCounts, partialAux, partialZ);
    finalize_kernel<<<1, 128, 0, stream>>>(
        blockCounts, blockOffsets, partialAux, partialZ, aux, z);
    scatter_kernel<<<NTOK / 256, 256, 0, stream>>>(
        expIdx, rnk, wgt, blockOffsets, dispatch, combine);
}